// RNNLayer_15032385536416
// MI455X (gfx1250) — compile-verified
//
#include <hip/hip_runtime.h>

#define N_BATCH 32
#define SEQ     2048
#define V_DIM   1024
#define KCHUNK  256
#define LDS_STRIDE (KCHUNK + 4)   // 260 floats: +4 pad -> bank-conflict-free frag reads

typedef float v2f __attribute__((ext_vector_type(2)));
typedef float v8f __attribute__((ext_vector_type(8)));

// One step of the recurrence: ys[:, t, :] = (xs[:, t, :] + h_prev) @ W^T + b
// Grid: 16 blocks x 128 threads (4 waves). Wave w of block b owns output
// columns [ (b*4+w)*16 , +16 ) and both 16-row M tiles (batch rows 0-31).
__global__ __launch_bounds__(128) void rnn_step_kernel(
    const float* __restrict__ xs,     // [N, S, V]
    const float* __restrict__ hprev,  // row n at hprev + n*h_stride
    long h_stride,
    const float* __restrict__ W,      // [V, V] row-major (out, in)
    const float* __restrict__ bias,   // [V]
    float* __restrict__ ys,           // [N, S, V]
    int t)
{
    __shared__ float Ash[N_BATCH * LDS_STRIDE];

    const int tid    = threadIdx.x;
    const int wave   = tid >> 5;
    const int lane   = tid & 31;
    const int lane16 = lane & 15;
    const int khalf  = (lane >> 4) << 1;      // 0 (lanes 0-15) or 2 (lanes 16-31)

    const int    n_base = (blockIdx.x * 4 + wave) * 16;
    const size_t SV     = (size_t)SEQ * V_DIM;
    const size_t tV     = (size_t)t * V_DIM;

    // Fuse bias: C/D layout has col n = lane&15 for every one of the 8 VGPRs.
    const float bn = bias[n_base + lane16];
    v8f acc0, acc1;
    #pragma unroll
    for (int g = 0; g < 8; ++g) { acc0[g] = bn; acc1[g] = bn; }

    for (int kc = 0; kc < V_DIM; kc += KCHUNK) {
        __syncthreads();
        // Cooperatively stage A-chunk = x_t + h_prev (32 rows x 256 k).
        // One float2 per thread covers a full row per iteration (coalesced 1KB);
        // all addressing is pointer increments (no per-element index math).
        {
            const float* xp = xs + tV + (size_t)kc + 2 * tid;
            const float* hp = hprev + (size_t)kc + 2 * tid;
            float*       lp = Ash + 2 * tid;
            #pragma unroll 8
            for (int m = 0; m < N_BATCH; ++m) {
                const v2f x = *(const v2f*)xp;
                const v2f h = *(const v2f*)hp;
                *(v2f*)lp = x + h;
                xp += SV;
                hp += h_stride;
                lp += LDS_STRIDE;
            }
        }
        __syncthreads();

        // K-loop: 4 per WMMA. B[k][n] = W[n][k]; lane supplies (k0+khalf, k0+khalf+1).
        const float* wrow = W + (size_t)(n_base + lane16) * V_DIM + (size_t)kc + khalf;
        #pragma unroll 4
        for (int kk = 0; kk < KCHUNK; kk += 4) {
            const v2f bfrag = *(const v2f*)(wrow + kk);
            const v2f a0 = *(const v2f*)(&Ash[ lane16       * LDS_STRIDE + kk + khalf]);
            const v2f a1 = *(const v2f*)(&Ash[(lane16 + 16) * LDS_STRIDE + kk + khalf]);
            acc0 = __builtin_amdgcn_wmma_f32_16x16x4_f32(
                false, a0, false, bfrag, (short)0, acc0, false, false);
            acc1 = __builtin_amdgcn_wmma_f32_16x16x4_f32(
                false, a1, false, bfrag, (short)0, acc1, false, false);
        }
    }

    // Store D: vgpr g holds (m = g + 8*(lane>=16), n = n_base + lane&15).
    // acc0 -> batch rows mhi..mhi+7, acc1 -> rows mhi+16..mhi+23.
    const int mhi = (lane >> 4) << 3;             // 0 or 8
    float* op0 = ys + (size_t)mhi * SV + tV + (n_base + lane16);
    float* op1 = op0 + (size_t)16 * SV;
    #pragma unroll
    for (int g = 0; g < 8; ++g) {
        *op0 = acc0[g];
        *op1 = acc1[g];
        op0 += SV;
        op1 += SV;
    }
}

// Copy ys[:, S-1, :] -> last output block.
__global__ void copy_last_kernel(const float* __restrict__ ys,
                                 float* __restrict__ out_last)
{
    const int i = blockIdx.x * blockDim.x + threadIdx.x;   // 0 .. N*V-1
    const int n = i >> 10;                                  // / V_DIM
    const int v = i & (V_DIM - 1);
    out_last[i] = ys[(size_t)n * SEQ * V_DIM + (size_t)(SEQ - 1) * V_DIM + v];
}

extern "C" void kernel_launch(void* const* d_in, const int* in_sizes, int n_in,
                              void* d_out, int out_size, void* d_ws, size_t ws_size,
                              hipStream_t stream)
{
    (void)in_sizes; (void)n_in; (void)out_size; (void)d_ws; (void)ws_size;

    const float* xs = (const float*)d_in[0];   // [N, S, V]
    const float* h0 = (const float*)d_in[1];   // [N, V]
    const float* W  = (const float*)d_in[2];   // [V, V]
    const float* b  = (const float*)d_in[3];   // [V]

    float* ys   = (float*)d_out;                                   // [N, S, V]
    float* last = ys + (size_t)N_BATCH * SEQ * V_DIM;              // [N, V]

    dim3 grid(16), block(128);
    for (int t = 0; t < SEQ; ++t) {
        // h_t lives in ys[:, t-1, :] for t>0 (row stride S*V), else in h0 (row stride V).
        const float* hp = (t == 0) ? h0 : (ys + (size_t)(t - 1) * V_DIM);
        const long   hs = (t == 0) ? (long)V_DIM : (long)SEQ * V_DIM;
        rnn_step_kernel<<<grid, block, 0, stream>>>(xs, hp, hs, W, b, ys, t);
    }
    copy_last_kernel<<<(N_BATCH * V_DIM) / 256, 256, 0, stream>>>(ys, last);
}